// ContinuousDeltaModel_35270271435181
// MI455X (gfx1250) — compile-verified
//
#include <hip/hip_runtime.h>
#include <hip/hip_bf16.h>
#include <stdint.h>

// Problem dims (fixed by reference)
#define BB 256
#define LL 4096
#define HH 32
#define VV 64

typedef __attribute__((ext_vector_type(2))) float v2f;
typedef __attribute__((ext_vector_type(8))) float v8f;
typedef int v4i_ __attribute__((vector_size(16)));  // matches builtin param type

// --- CDNA5 f32 WMMA: D(16x16) = A(16x4) * B(4x16) + C ------------------
__device__ __forceinline__ v8f wmma_f32_16x16x4(v2f a, v2f b, v8f c) {
  return __builtin_amdgcn_wmma_f32_16x16x4_f32(
      /*neg_a=*/false, a, /*neg_b=*/false, b,
      /*c_mod=*/(short)0, c, /*reuse_a=*/false, /*reuse_b=*/false);
}

// --- CDNA5 async global->LDS copy (guarded; falls back to plain copy) ---
#if __has_builtin(__builtin_amdgcn_global_load_async_to_lds_b128) && \
    __has_builtin(__builtin_amdgcn_s_wait_asynccnt)
#define HAVE_ASYNC_LDS 1
#endif

__device__ __forceinline__ void cp16_g2l(const float* g, float* l) {
#ifdef HAVE_ASYNC_LDS
  __builtin_amdgcn_global_load_async_to_lds_b128(
      (__attribute__((address_space(1))) v4i_*)((v4i_*)g),
      (__attribute__((address_space(3))) v4i_*)((v4i_*)l),
      /*offset=*/0, /*cpol=*/0);
#else
  *(float4*)l = *(const float4*)g;
#endif
}

__device__ __forceinline__ void wait_async_zero() {
#ifdef HAVE_ASYNC_LDS
  __builtin_amdgcn_s_wait_asynccnt(0);
#endif
}

// ========================================================================
// Kernel 1: encode. One wave handles 16 tokens with f32 WMMA.
//   h = embed[seq];  ff = relu(h@w1+b1)@w2+b2;  x = h+ff;  LN -> hidden
// Fragment layouts per ISA 7.12.2:
//   A 16x4 : lane L holds (m=L%16, K=2*(L/16) .. +1)  -> v2f
//   B 4x16 : lane L holds (n=L%16, K=2*(L/16) .. +1)  -> v2f
//   C 16x16: VGPR r, lanes0-15 -> row r, lanes16-31 -> row r+8; col = L%16
// ========================================================================
__global__ __launch_bounds__(256) void encode_kernel(
    const int* __restrict__ seq,      // BB*LL
    const float* __restrict__ embed,  // VV*HH
    const float* __restrict__ w1,     // 32*64
    const float* __restrict__ b1,     // 64
    const float* __restrict__ w2,     // 64*32
    const float* __restrict__ b2,     // 32
    const float* __restrict__ ln_g,   // 32
    const float* __restrict__ ln_b,   // 32
    float* __restrict__ hidden)       // BB*LL*HH
{
  __shared__ __align__(16) float s_w1[32 * 64];
  __shared__ __align__(16) float s_w2[64 * 32];
  __shared__ __align__(16) float s_b1[64];
  __shared__ __align__(16) float s_b2[32], s_g[32], s_bl[32];
  __shared__ __align__(16) float s_h[8][16 * 32];   // per-wave h tile
  __shared__ __align__(16) float s_a1[8][16 * 64];  // per-wave relu(h@w1+b1)
  __shared__ __align__(16) float s_x[8][16 * 32];   // per-wave residual out

  const int tid = threadIdx.x;

  // ---- stage weights via async DMA to LDS (16B per lane per round) ----
  for (int i = tid; i < (32 * 64) / 4; i += 256) cp16_g2l(w1 + i * 4, s_w1 + i * 4);
  for (int i = tid; i < (64 * 32) / 4; i += 256) cp16_g2l(w2 + i * 4, s_w2 + i * 4);
  if (tid < 64) s_b1[tid] = b1[tid];
  if (tid < 32) { s_b2[tid] = b2[tid]; s_g[tid] = ln_g[tid]; s_bl[tid] = ln_b[tid]; }
  wait_async_zero();
  __syncthreads();

  const int wave = tid >> 5, lane = tid & 31;
  const int half = lane >> 4, l16 = lane & 15;
  const long tokBase = ((long)blockIdx.x * 8 + wave) * 16;

  // ---- gather embeddings: lane pair (l16, l16+16) covers one token ----
  {
    const int idx = seq[tokBase + l16];
    const float* er = embed + idx * HH + half * 16;
    float* dr = &s_h[wave][l16 * 32 + half * 16];
#pragma unroll
    for (int i = 0; i < 4; ++i)
      *(float4*)(dr + i * 4) = *(const float4*)(er + i * 4);
  }
  __syncthreads();

  // ---- A fragments for layer 1 (K = 32 -> 8 k-steps) ----
  v2f afrag[8];
  {
    const float* hr = &s_h[wave][l16 * 32];
#pragma unroll
    for (int ks = 0; ks < 8; ++ks)
      afrag[ks] = *(const v2f*)(hr + ks * 4 + 2 * half);
  }

  // ---- layer 1: (16x32)@(32x64), 4 N-tiles x 8 K-steps = 32 WMMA ----
#pragma unroll
  for (int nt = 0; nt < 4; ++nt) {
    v8f acc = {0.f, 0.f, 0.f, 0.f, 0.f, 0.f, 0.f, 0.f};
#pragma unroll
    for (int ks = 0; ks < 8; ++ks) {
      const int kr = ks * 4 + 2 * half;
      v2f bf;
      bf.x = s_w1[kr * 64 + nt * 16 + l16];
      bf.y = s_w1[(kr + 1) * 64 + nt * 16 + l16];
      acc = wmma_f32_16x16x4(afrag[ks], bf, acc);
    }
    const float bias = s_b1[nt * 16 + l16];
#pragma unroll
    for (int r = 0; r < 8; ++r) {
      const float v = fmaxf(acc[r] + bias, 0.f);
      s_a1[wave][(r + 8 * half) * 64 + nt * 16 + l16] = v;  // C-layout -> LDS
    }
  }
  __syncthreads();

  // ---- layer 2: (16x64)@(64x32), 2 N-tiles x 16 K-steps = 32 WMMA ----
#pragma unroll
  for (int nt = 0; nt < 2; ++nt) {
    v8f acc = {0.f, 0.f, 0.f, 0.f, 0.f, 0.f, 0.f, 0.f};
#pragma unroll
    for (int ks = 0; ks < 16; ++ks) {
      const int kr = ks * 4 + 2 * half;
      v2f af = *(const v2f*)(&s_a1[wave][l16 * 64 + kr]);
      v2f bf;
      bf.x = s_w2[kr * 32 + nt * 16 + l16];
      bf.y = s_w2[(kr + 1) * 32 + nt * 16 + l16];
      acc = wmma_f32_16x16x4(af, bf, acc);
    }
    const float bias = s_b2[nt * 16 + l16];
#pragma unroll
    for (int r = 0; r < 8; ++r) {
      const int m = r + 8 * half;
      const float xv = acc[r] + bias + s_h[wave][m * 32 + nt * 16 + l16];
      s_x[wave][m * 32 + nt * 16 + l16] = xv;
    }
  }
  __syncthreads();

  // ---- LayerNorm: lane pair (l16, l16+16) reduces token l16's 32 dims ----
  {
    const float* xr = &s_x[wave][l16 * 32 + half * 16];
    float4 xv[4];
    float sum = 0.f, sq = 0.f;
#pragma unroll
    for (int i = 0; i < 4; ++i) {
      xv[i] = *(const float4*)(xr + i * 4);
      sum += xv[i].x + xv[i].y + xv[i].z + xv[i].w;
      sq += xv[i].x * xv[i].x + xv[i].y * xv[i].y + xv[i].z * xv[i].z + xv[i].w * xv[i].w;
    }
    sum += __shfl_xor(sum, 16, 32);
    sq  += __shfl_xor(sq, 16, 32);
    const float mu = sum * (1.f / 32.f);
    const float var = sq * (1.f / 32.f) - mu * mu;
    const float rs = rsqrtf(var + 1e-5f);
    float* out = hidden + (tokBase + l16) * HH + half * 16;
#pragma unroll
    for (int i = 0; i < 4; ++i) {
      const int d = half * 16 + i * 4;
      float4 y;
      y.x = (xv[i].x - mu) * rs * s_g[d + 0] + s_bl[d + 0];
      y.y = (xv[i].y - mu) * rs * s_g[d + 1] + s_bl[d + 1];
      y.z = (xv[i].z - mu) * rs * s_g[d + 2] + s_bl[d + 2];
      y.w = (xv[i].w - mu) * rs * s_g[d + 3] + s_bl[d + 3];
      *(float4*)(out + i * 4) = y;
    }
  }
}

// ========================================================================
// Kernel 2: delta scan. One wave per batch. Lane i owns row i of M.
// k is wave-uniform -> scalar (SMEM) loads; per step ~100 VALU ops, no
// cross-lane traffic. Also produces context = M @ hidden[b, L-1].
// ========================================================================
__global__ __launch_bounds__(32) void delta_scan_kernel(
    const float* __restrict__ hidden,  // BB*LL*HH
    float* __restrict__ ctx)           // BB*HH
{
  const int b = blockIdx.x;
  const int lane = threadIdx.x;
  const float* base = hidden + (size_t)b * LL * HH;

  float m[HH];
#pragma unroll
  for (int j = 0; j < HH; ++j) m[j] = 0.f;

  for (int t = 0; t < LL - 1; ++t) {
    const float* kp = base + t * HH;
    __builtin_prefetch(kp + 16 * HH, 0, 1);  // stream k's ahead (global_prefetch)

    float k[HH];
#pragma unroll
    for (int j = 0; j < HH; j += 4) {
      const float4 v = *(const float4*)(kp + j);  // uniform -> s_load
      k[j] = v.x; k[j + 1] = v.y; k[j + 2] = v.z; k[j + 3] = v.w;
    }
    const float kv = kp[lane];  // per-lane k[i]

    float vp0 = 0.f, vp1 = 0.f, vp2 = 0.f, vp3 = 0.f, dn = 0.f;
#pragma unroll
    for (int j = 0; j < HH; j += 4) {
      vp0 = fmaf(m[j + 0], k[j + 0], vp0);
      vp1 = fmaf(m[j + 1], k[j + 1], vp1);
      vp2 = fmaf(m[j + 2], k[j + 2], vp2);
      vp3 = fmaf(m[j + 3], k[j + 3], vp3);
    }
#pragma unroll
    for (int j = 0; j < HH; ++j) dn = fmaf(k[j], k[j], dn);

    const float vp = (vp0 + vp1) + (vp2 + vp3);
    const float delta = kv - vp / (dn + 1e-6f);
#pragma unroll
    for (int j = 0; j < HH; ++j) m[j] = fmaf(delta, k[j], m[j]);
  }

  // context = M @ query (query = last token's hidden)
  const float* qp = base + (size_t)(LL - 1) * HH;
  float c0 = 0.f, c1 = 0.f, c2 = 0.f, c3 = 0.f;
#pragma unroll
  for (int j = 0; j < HH; j += 4) {
    const float4 v = *(const float4*)(qp + j);
    c0 = fmaf(m[j + 0], v.x, c0);
    c1 = fmaf(m[j + 1], v.y, c1);
    c2 = fmaf(m[j + 2], v.z, c2);
    c3 = fmaf(m[j + 3], v.w, c3);
  }
  ctx[b * HH + lane] = (c0 + c1) + (c2 + c3);
}

// ========================================================================
// Kernel 3: logits = ctx @ (rp_w@out_w) + (rp_b@out_w + out_b), via WMMA.
// Single block: fold the two projections into one 32x64 matrix in LDS,
// then 256x64x32 GEMM = 16 M-tiles x 4 N-tiles x 8 K-steps of f32 WMMA.
// ========================================================================
__global__ __launch_bounds__(256) void proj_kernel(
    const float* __restrict__ ctx,    // 256*32
    const float* __restrict__ rp_w,   // 32*32
    const float* __restrict__ rp_b,   // 32
    const float* __restrict__ out_w,  // 32*64
    const float* __restrict__ out_b,  // 64
    float* __restrict__ logits)       // 256*64
{
  __shared__ __align__(16) float s_W[32 * 64];
  __shared__ __align__(16) float s_bb[64];
  __shared__ __align__(16) float s_c[256 * 32];

  const int tid = threadIdx.x;

  // Wf[i][n] = sum_j rp_w[i][j] * out_w[j][n]
  for (int o = tid; o < 32 * 64; o += 256) {
    const int i = o >> 6, n = o & 63;
    float acc = 0.f;
#pragma unroll
    for (int j = 0; j < 32; ++j) acc = fmaf(rp_w[i * 32 + j], out_w[j * 64 + n], acc);
    s_W[o] = acc;
  }
  if (tid < 64) {
    float acc = out_b[tid];
#pragma unroll
    for (int j = 0; j < 32; ++j) acc = fmaf(rp_b[j], out_w[j * 64 + tid], acc);
    s_bb[tid] = acc;
  }
  for (int i = tid; i < (256 * 32) / 4; i += 256)
    *(float4*)(s_c + i * 4) = *(const float4*)(ctx + i * 4);
  __syncthreads();

  const int wave = tid >> 5, lane = tid & 31;
  const int half = lane >> 4, l16 = lane & 15;

  for (int mt = wave; mt < 16; mt += 8) {
    v2f af[8];
#pragma unroll
    for (int ks = 0; ks < 8; ++ks)
      af[ks] = *(const v2f*)(&s_c[(mt * 16 + l16) * 32 + ks * 4 + 2 * half]);
#pragma unroll
    for (int nt = 0; nt < 4; ++nt) {
      v8f acc = {0.f, 0.f, 0.f, 0.f, 0.f, 0.f, 0.f, 0.f};
#pragma unroll
      for (int ks = 0; ks < 8; ++ks) {
        const int kr = ks * 4 + 2 * half;
        v2f bf;
        bf.x = s_W[kr * 64 + nt * 16 + l16];
        bf.y = s_W[(kr + 1) * 64 + nt * 16 + l16];
        acc = wmma_f32_16x16x4(af[ks], bf, acc);
      }
      const float bias = s_bb[nt * 16 + l16];
#pragma unroll
      for (int r = 0; r < 8; ++r)
        logits[(mt * 16 + r + 8 * half) * 64 + nt * 16 + l16] = acc[r] + bias;
    }
  }
}

// ========================================================================
extern "C" void kernel_launch(void* const* d_in, const int* in_sizes, int n_in,
                              void* d_out, int out_size, void* d_ws, size_t ws_size,
                              hipStream_t stream) {
  const int*   seq   = (const int*)d_in[0];
  const float* embed = (const float*)d_in[1];
  const float* w1    = (const float*)d_in[2];
  const float* b1    = (const float*)d_in[3];
  const float* w2    = (const float*)d_in[4];
  const float* b2    = (const float*)d_in[5];
  const float* ln_g  = (const float*)d_in[6];
  const float* ln_b  = (const float*)d_in[7];
  const float* rp_w  = (const float*)d_in[8];
  const float* rp_b  = (const float*)d_in[9];
  const float* out_w = (const float*)d_in[10];
  const float* out_b = (const float*)d_in[11];
  float* logits = (float*)d_out;

  const size_t hiddenElems = (size_t)BB * LL * HH;  // 128 MB fp32
  const size_t needed = (hiddenElems + (size_t)BB * HH) * sizeof(float);
  if (ws_size < needed) return;  // workspace too small; nothing safe to do

  float* hidden = (float*)d_ws;
  float* ctx = hidden + hiddenElems;

  // 1M tokens / (8 waves * 16 tokens) = 8192 blocks
  encode_kernel<<<dim3((BB * LL) / 128), dim3(256), 0, stream>>>(
      seq, embed, w1, b1, w2, b2, ln_g, ln_b, hidden);

  // one wave per batch chain
  delta_scan_kernel<<<dim3(BB), dim3(32), 0, stream>>>(hidden, ctx);

  // single-block fused double projection
  proj_kernel<<<dim3(1), dim3(256), 0, stream>>>(ctx, rp_w, rp_b, out_w, out_b, logits);
}